// HarmonicSynth_28303834481108
// MI455X (gfx1250) — compile-verified
//
#include <hip/hip_runtime.h>
#include <hip/hip_bf16.h>
#include <math.h>

#define PI_F 3.14159265358979323846f
#define SR_F 16000.0f
#define HOP_F 64.0f
#define NYQ_F (16000.0f * 0.45f)
#define NH 64            // harmonics (problem shape)
#define TILE 1024        // samples per workgroup in synth
#define MAXFR 20         // staged frames per tile (need 19)

typedef __attribute__((ext_vector_type(2))) float v2f;
typedef __attribute__((ext_vector_type(8))) float v8f;

// ---------------------------------------------------------------------------
// Kernel 1: per-batch scalars: f0_mean, c_k = k*stretch, sigma_k, 1/n_active
// ---------------------------------------------------------------------------
__global__ void __launch_bounds__(256)
prep_kernel(const float* __restrict__ f0, const float* __restrict__ inh,
            const float* __restrict__ b1, const float* __restrict__ b3,
            float* __restrict__ c_ws, float* __restrict__ sig_ws,
            float* __restrict__ inv_na, int T, int N) {
  int b = blockIdx.x, tid = threadIdx.x;
  __shared__ float red[256];
  float s = 0.f;
  for (int t = tid; t < T; t += 256) s += fmaxf(f0[b * T + t], 20.f);
  red[tid] = s;
  __syncthreads();
  for (int o = 128; o > 0; o >>= 1) {
    if (tid < o) red[tid] += red[tid + o];
    __syncthreads();
  }
  float f0m = red[0] / (float)T;
  if (tid < N) {
    float k = (float)(tid + 1);
    float stretch = sqrtf(1.f + inh[b] * k * k);
    c_ws[b * N + tid] = k * stretch;
    float omega = 2.f * PI_F * f0m * k;
    sig_ws[b * N + tid] = b1[b] + b3[b] * omega * omega;
  }
  if (tid == 0) {
    int na = 0;
    for (int i = 1; i <= N; ++i)
      if ((float)i * f0m < NYQ_F) na++;
    if (na < 1) na = 1;
    inv_na[b] = 1.f / (float)na;
  }
}

// ---------------------------------------------------------------------------
// Kernel 2: a_dec[b][t][n] = harm[b][t][n] * exp(-sigma[b][n] * t*hop/SR)
// (pure elementwise in the input layout -> fully coalesced, v_exp_f32)
// ---------------------------------------------------------------------------
__global__ void __launch_bounds__(256)
decay_kernel(const float* __restrict__ harm, const float* __restrict__ sig_ws,
             float* __restrict__ a_dec, int B, int T, int N) {
  int idx = blockIdx.x * 256 + threadIdx.x;
  int tot = B * T * N;
  if (idx >= tot) return;
  int n = idx % N;
  int bt = idx / N;
  int t = bt % T;
  int b = bt / T;
  float tt = (float)t * (HOP_F / SR_F);
  a_dec[idx] = harm[idx] * __expf(-sig_ws[b * N + n] * tt);
}

// ---------------------------------------------------------------------------
// Kernel 3: S[b][j] = inclusive cumsum_j of f0_up(b,j)/SR.
// One 1024-thread workgroup per batch; wave-shuffle hierarchical scan.
// ---------------------------------------------------------------------------
__global__ void __launch_bounds__(1024)
scan_kernel(const float* __restrict__ f0, float* __restrict__ S, int T, int nsamp) {
  int b = blockIdx.x, tid = threadIdx.x;
  __shared__ float f0sh[1024];
  __shared__ float wagg[32];
  for (int t = tid; t < T; t += 1024) f0sh[t] = f0[b * T + t];
  __syncthreads();

  int CH = (nsamp + 1023) >> 10;
  int j0 = tid * CH;
  int j1 = j0 + CH; if (j1 > nsamp) j1 = nsamp;
  float scale = (float)T / (float)nsamp;
  const float inv_sr = 1.f / SR_F;

  float s = 0.f;
  for (int j = j0; j < j1; ++j) {
    float pos = ((float)j + 0.5f) * scale - 0.5f;
    pos = fminf(fmaxf(pos, 0.f), (float)(T - 1));
    int lo = (int)pos;
    int hi = lo + 1; if (hi > T - 1) hi = T - 1;
    float w = pos - (float)lo;
    s += (f0sh[lo] + (f0sh[hi] - f0sh[lo]) * w) * inv_sr;
  }
  int lane = tid & 31;
  float v = s;
  for (int o = 1; o < 32; o <<= 1) {
    float t2 = __shfl_up(v, o, 32);
    if (lane >= o) v += t2;
  }
  if (lane == 31) wagg[tid >> 5] = v;
  __syncthreads();
  if (tid < 32) {
    float x = wagg[tid], inc = x;
    for (int o = 1; o < 32; o <<= 1) {
      float t2 = __shfl_up(inc, o, 32);
      if (tid >= o) inc += t2;
    }
    wagg[tid] = inc - x;  // exclusive wave offset
  }
  __syncthreads();
  float run = (v - s) + wagg[tid >> 5];
  for (int j = j0; j < j1; ++j) {
    float pos = ((float)j + 0.5f) * scale - 0.5f;
    pos = fminf(fmaxf(pos, 0.f), (float)(T - 1));
    int lo = (int)pos;
    int hi = lo + 1; if (hi > T - 1) hi = T - 1;
    float w = pos - (float)lo;
    run += (f0sh[lo] + (f0sh[hi] - f0sh[lo]) * w) * inv_sr;
    S[(size_t)b * nsamp + j] = run;
  }
}

// ---------------------------------------------------------------------------
// Kernel 4: synthesis. Wave handles 16-sample columns; harmonic sum via
// V_WMMA_F32_16X16X4_F32 with all-ones A (column-sum reduction on matrix pipe).
// B-frag lane mapping (ISA 32-bit A/B layout): VGPR0 -> K = (lane<16?0:2),
// VGPR1 -> K+1, column N = lane%16. Column sums are K-permutation invariant.
// ---------------------------------------------------------------------------
__global__ void __launch_bounds__(256)
synth_kernel(const float* __restrict__ a_dec, const float* __restrict__ S,
             const float* __restrict__ f0, const float* __restrict__ c_ws,
             const float* __restrict__ inv_na, float* __restrict__ out,
             int T, int nsamp) {
  int b = blockIdx.y;
  int tile0 = blockIdx.x * TILE;
  int tid = threadIdx.x;

  __shared__ float ldsA[MAXFR * NH];
  __shared__ float f0s[MAXFR];
  __shared__ float cc[NH];

  float scale = (float)T / (float)nsamp;  // 0.015625 for this problem
  int fbase = (int)floorf(((float)tile0 + 0.5f) * scale - 0.5f);
  if (fbase < 0) fbase = 0;
  int FR = (int)((float)TILE * scale) + 3;  // 19 frames staged
  if (FR > MAXFR) FR = MAXFR;

  // Stage decayed-amplitude slice (frame-major, rows contiguous) and f0 slice.
  for (int i = tid; i < FR * NH; i += 256) {
    int fr = i >> 6, hh = i & 63;
    int f = fbase + fr;
    if (f < T) ldsA[i] = a_dec[((size_t)b * T + f) * NH + hh];
  }
  for (int i = tid; i < FR; i += 256) {
    int f = fbase + i;
    f0s[i] = (f < T) ? f0[b * T + f] : 0.f;
  }
  if (tid < NH) cc[tid] = c_ws[b * NH + tid];
  __syncthreads();

  float inv = inv_na[b];
  int w = tid >> 5, l = tid & 31;
  int ncol = l & 15;
  bool hiHalf = (l >= 16);
  v2f ones; ones.x = 1.f; ones.y = 1.f;

  for (int st = w; st < TILE / 16; st += 8) {
    int j = tile0 + st * 16 + ncol;
    bool jv = (j < nsamp);
    int jc = jv ? j : nsamp - 1;

    float pos = ((float)jc + 0.5f) * scale - 0.5f;
    pos = fminf(fmaxf(pos, 0.f), (float)(T - 1));
    int lo = (int)pos;
    int hi = lo + 1; if (hi > T - 1) hi = T - 1;
    float wt = pos - (float)lo;
    int rl = lo - fbase, rh = hi - fbase;

    float f0u = f0s[rl] + (f0s[rh] - f0s[rl]) * wt;
    float Sv = S[(size_t)b * nsamp + jc];

    // software prefetch of the next tile's cumsum data (global_prefetch_b8)
    int jn = jc + TILE; if (jn > nsamp - 1) jn = nsamp - 1;
    __builtin_prefetch(S + (size_t)b * nsamp + jn, 0, 1);

    v8f acc = {0.f, 0.f, 0.f, 0.f, 0.f, 0.f, 0.f, 0.f};
#pragma unroll
    for (int kk = 0; kk < NH / 4; ++kk) {
      int h0 = kk * 4 + (hiHalf ? 2 : 0);
      int h1 = h0 + 1;
      float c0 = cc[h0], c1 = cc[h1];
      float a0 = ldsA[rl * NH + h0];
      a0 += (ldsA[rh * NH + h0] - a0) * wt;
      float a1 = ldsA[rl * NH + h1];
      a1 += (ldsA[rh * NH + h1] - a1) * wt;
      // phase in "turns": v_sin_f32 computes sin(2*pi*x); fract keeps accuracy
      float u0 = c0 * Sv; u0 -= floorf(u0);
      float u1 = c1 * Sv; u1 -= floorf(u1);
      float s0 = __builtin_amdgcn_sinf(u0);
      float s1 = __builtin_amdgcn_sinf(u1);
      float p0 = (jv && (f0u * c0 < NYQ_F)) ? a0 * s0 : 0.f;
      float p1 = (jv && (f0u * c1 < NYQ_F)) ? a1 * s1 : 0.f;
      v2f bf; bf.x = p0; bf.y = p1;
      // D = ones(16x4) x P(4x16) + D  -> column sums accumulate over harmonics
      acc = __builtin_amdgcn_wmma_f32_16x16x4_f32(
          false, ones, false, bf, (short)0, acc, false, false);
    }
    // every lane's acc[0] holds signal for column n = lane%16 (rows identical)
    if (!hiHalf && jv) out[(size_t)b * nsamp + j] = acc[0] * inv;
  }
}

// ---------------------------------------------------------------------------
extern "C" void kernel_launch(void* const* d_in, const int* in_sizes, int n_in,
                              void* d_out, int out_size, void* d_ws, size_t ws_size,
                              hipStream_t stream) {
  const float* harm = (const float*)d_in[0];
  const float* f0   = (const float*)d_in[1];
  const float* inh  = (const float*)d_in[3];
  const float* b1   = (const float*)d_in[4];
  const float* b3   = (const float*)d_in[5];
  float* out = (float*)d_out;

  int B = in_sizes[3];                 // inh is (B,)
  int T = in_sizes[1] / B;             // f0 is (B,T)
  int N = in_sizes[0] / (B * T);       // harm is (B,T,N); N == 64
  int nsamp = out_size / B;            // output (B,1,n)

  float* a_dec  = (float*)d_ws;                       // B*T*N
  float* S      = a_dec + (size_t)B * T * N;          // B*nsamp
  float* c_ws   = S + (size_t)B * nsamp;              // B*N
  float* sig_ws = c_ws + (size_t)B * N;               // B*N
  float* inv_na = sig_ws + (size_t)B * N;             // B

  prep_kernel<<<B, 256, 0, stream>>>(f0, inh, b1, b3, c_ws, sig_ws, inv_na, T, N);

  int tot = B * T * N;
  decay_kernel<<<(tot + 255) / 256, 256, 0, stream>>>(harm, sig_ws, a_dec, B, T, N);

  scan_kernel<<<B, 1024, 0, stream>>>(f0, S, T, nsamp);

  int tiles = (nsamp + TILE - 1) / TILE;
  dim3 grid(tiles, B);
  synth_kernel<<<grid, 256, 0, stream>>>(a_dec, S, f0, c_ws, inv_na, out, T, nsamp);
}